// SocialAttention_28381143892377
// MI455X (gfx1250) — compile-verified
//
#include <hip/hip_runtime.h>
#include <math.h>

typedef __attribute__((ext_vector_type(2))) float v2f;
typedef __attribute__((ext_vector_type(8))) float v8f;

#define S_    64
#define NPS   48
#define NROWS (S_*NPS)   // 3072
#define H_    256
#define A_    64

// ---------------------------------------------------------------------------
// Kernel 1: proj_t[3072][64] = temporal[3072][256] @ W2_w^T + W2_b
// One 16x16 output tile per wave via V_WMMA_F32_16X16X4_F32, K accumulated
// in steps of 4.  Tiles: M=192, N=4 -> 768 waves -> 96 blocks of 8 waves.
// ---------------------------------------------------------------------------
__global__ __launch_bounds__(256) void k_proj_t(
    const float* __restrict__ T, const float* __restrict__ W2w,
    const float* __restrict__ W2b, float* __restrict__ P)
{
  const int tid  = threadIdx.x;
  const int lane = tid & 31;
  const int wid  = tid >> 5;
  const int gw   = blockIdx.x * 8 + wid;   // 0..767
  const int tileM = gw >> 2;               // 0..191
  const int tileN = gw & 3;                // 0..3
  const int half  = lane >> 4;             // K-pair select
  const int l16   = lane & 15;             // M row (A) / N col (B)

  const float* arow = T   + (size_t)(tileM*16 + l16) * H_;   // A[m][k]
  const float* brow = W2w + (size_t)(tileN*16 + l16) * H_;   // B[k][n] = W2_w[n][k]

  v8f acc = {0.f,0.f,0.f,0.f,0.f,0.f,0.f,0.f};
  for (int k0 = 0; k0 < H_; k0 += 4) {
    const int ka = k0 + 2*half;
    v2f a, b;
    a.x = arow[ka];  a.y = arow[ka + 1];
    b.x = brow[ka];  b.y = brow[ka + 1];
    acc = __builtin_amdgcn_wmma_f32_16x16x4_f32(
        /*neg_a=*/false, a, /*neg_b=*/false, b,
        /*c_mod=*/(short)0, acc, /*reuse_a=*/false, /*reuse_b=*/false);
  }
  const int ncol  = tileN*16 + l16;
  const float bias = W2b[ncol];
  #pragma unroll
  for (int v = 0; v < 8; ++v) {
    const int mrow = tileM*16 + v + 8*half;       // C/D layout: VGPR v, half
    P[(size_t)mrow*A_ + ncol] = acc[v] + bias;
  }
}

// ---------------------------------------------------------------------------
// Kernel 2: V[3072][256] = proj_t[3072][64] @ W1_w[64][256]
// Tiles: M=192, N=16 -> 3072 waves -> 384 blocks of 8 waves. K=64.
// ---------------------------------------------------------------------------
__global__ __launch_bounds__(256) void k_vproj(
    const float* __restrict__ P, const float* __restrict__ W1w,
    float* __restrict__ V)
{
  const int tid  = threadIdx.x;
  const int lane = tid & 31;
  const int wid  = tid >> 5;
  const int gw   = blockIdx.x * 8 + wid;   // 0..3071
  const int tileM = gw >> 4;               // 0..191
  const int tileN = gw & 15;               // 0..15
  const int half  = lane >> 4;
  const int l16   = lane & 15;

  const float* arow = P + (size_t)(tileM*16 + l16) * A_;
  const int ncol = tileN*16 + l16;

  v8f acc = {0.f,0.f,0.f,0.f,0.f,0.f,0.f,0.f};
  for (int k0 = 0; k0 < A_; k0 += 4) {
    const int ka = k0 + 2*half;
    v2f a, b;
    a.x = arow[ka];            a.y = arow[ka + 1];
    b.x = W1w[(size_t)ka*H_       + ncol];       // B[k][h] = W1_w[a][h]
    b.y = W1w[(size_t)(ka+1)*H_   + ncol];
    acc = __builtin_amdgcn_wmma_f32_16x16x4_f32(
        false, a, false, b, (short)0, acc, false, false);
  }
  #pragma unroll
  for (int v = 0; v < 8; ++v) {
    const int mrow = tileM*16 + v + 8*half;
    V[(size_t)mrow*H_ + ncol] = acc[v];
  }
}

// ---------------------------------------------------------------------------
// Kernel 3: fused score + masked exp + normalize + edge aggregation.
// One workgroup (8 wave32s) per (s,i) row.  The 48x256 f32 tile (48 KB) is
// DMA'd into LDS once via GLOBAL_LOAD_ASYNC_TO_LDS_B128 (ASYNCcnt path --
// no VGPR round-trip), then consumed by both the score matvec and the
// aggregation.  Single HBM pass over spatial_ht (151 MB) -> bandwidth bound.
// ---------------------------------------------------------------------------
__global__ __launch_bounds__(256) void k_attn(
    const float* __restrict__ X,       // spatial_ht [S,48,48,256]
    const float* __restrict__ V,       // [3072,256]
    const float* __restrict__ P,       // [3072,64]
    const float* __restrict__ W1b,     // [64]
    const int*   __restrict__ tsmask,  // [3072]
    float* __restrict__ out)           // [3072,256]
{
  __shared__ float xs[NPS][H_];        // 49152 B
  __shared__ float vs[H_];
  __shared__ float attn[NPS];
  __shared__ int   msk[NPS];
  __shared__ float scal[3];            // [0]=ne, [1]=c, [2]=1/den

  const int tid = threadIdx.x;
  const int row = blockIdx.x;          // 0..3071
  const int s   = row / NPS;
  const int i   = row - s * NPS;

  // Stage tile: 12288 floats = 3072 x 16B, 12 async B128 copies per thread.
  // Generic LDS pointer low 32 bits == wave-relative LDS byte address
  // (ISA 10.2 aperture mapping), which is exactly what VDST wants.
  {
    const unsigned long long gbase =
        (unsigned long long)(const void*)(X + (size_t)row * NPS * H_);
    const unsigned lbase = (unsigned)(unsigned long long)(const void*)&xs[0][0];
    #pragma unroll
    for (int k = 0; k < (NPS*H_*4/16)/256; ++k) {       // 12
      const unsigned byteoff = (unsigned)((k*256 + tid) * 16);
      const unsigned long long ga = gbase + byteoff;
      const unsigned           la = lbase + byteoff;
      asm volatile("global_load_async_to_lds_b128 %0, %1, off"
                   :: "v"(la), "v"(ga) : "memory");
    }
  }

  vs[tid] = V[(size_t)row * H_ + tid];
  if (tid < NPS) msk[tid] = tsmask[s*NPS + tid];
  if (tid == 0) {
    float c = 0.f;
    #pragma unroll 8
    for (int a = 0; a < A_; ++a) c += P[(size_t)row*A_ + a] * W1b[a];
    scal[1] = c;
  }

  // All async DMA into xs[] must land before anyone reads it.
  asm volatile("s_wait_asynccnt 0x0" ::: "memory");
  __syncthreads();

  if (tid == 0) {
    int ne = 0;
    #pragma unroll
    for (int j = 0; j < NPS; ++j) ne += (msk[j] == 1);
    scal[0] = (float)ne;
  }
  __syncthreads();

  const float ne    = scal[0];
  const float c     = scal[1];
  const float scale = ne * 0.125f;         // ne / sqrt(A=64)
  const bool  mi    = (msk[i] == 1);

  // Scores: wave w handles rows j = w, w+8, ... (6 rows/wave).
  const int wid  = tid >> 5;
  const int lane = tid & 31;
  for (int j = wid; j < NPS; j += 8) {
    float p = 0.f;
    #pragma unroll
    for (int h = lane; h < H_; h += 32) p += vs[h] * xs[j][h];
    #pragma unroll
    for (int off = 16; off > 0; off >>= 1) p += __shfl_down(p, off, 32);
    if (lane == 0) {
      const bool valid = mi && (msk[j] == 1) && (j != i);
      const float sc = (p + c) * scale;
      attn[j] = valid ? __expf(sc) : 0.f;  // reference: unstabilized exp
    }
  }
  __syncthreads();

  if (tid == 0) {
    float den = 0.f;
    #pragma unroll
    for (int j = 0; j < NPS; ++j) den += attn[j];
    scal[2] = 1.0f / (den > 0.f ? den : 1.0f);
  }
  __syncthreads();

  // Aggregation: thread tid owns column h = tid; conflict-free LDS reads.
  const float invden = scal[2];
  float acc = 0.f;
  #pragma unroll
  for (int j = 0; j < NPS; ++j) acc += (attn[j] * invden) * xs[j][tid];
  out[(size_t)row * H_ + tid] = acc;
}

// ---------------------------------------------------------------------------
extern "C" void kernel_launch(void* const* d_in, const int* in_sizes, int n_in,
                              void* d_out, int out_size, void* d_ws, size_t ws_size,
                              hipStream_t stream) {
  (void)in_sizes; (void)n_in; (void)out_size; (void)ws_size;
  const float* X   = (const float*)d_in[0];  // spatial_ht
  const float* T   = (const float*)d_in[1];  // temporal_ht
  const float* W1w = (const float*)d_in[2];
  const float* W1b = (const float*)d_in[3];
  const float* W2w = (const float*)d_in[4];
  const float* W2b = (const float*)d_in[5];
  const int*   tm  = (const int*)d_in[6];
  float* out = (float*)d_out;

  float* P = (float*)d_ws;                   // [3072 x 64]
  float* V = P + (size_t)NROWS * A_;         // [3072 x 256]

  k_proj_t<<<96,    256, 0, stream>>>(T, W2w, W2b, P);
  k_vproj <<<384,   256, 0, stream>>>(P, W1w, V);
  k_attn  <<<NROWS, 256, 0, stream>>>(X, V, P, W1b, tm, out);
}